// ASIS_16836271801026
// MI455X (gfx1250) — compile-verified
//
#include <hip/hip_runtime.h>
#include <hip/hip_bf16.h>

// ---------------- problem constants (match reference) ----------------
#define BATCH   4
#define NPTS    4096
#define CCH     128      // C_SEM == C_INS == 128
#define SEMO    13
#define INSO    5
#define KNN     30
#define TN      128      // points per workgroup tile (K1/K3)

// Fragment-packed LDS layouts:
//  for k-group k0 (4 consecutive K values) and row/point r:
//    buf[k0*GS + r*4 + (k&3)]
//  -> lane (lm, lh) fragment = b64 at  k0*GS + r*4 + 2*lh
#define KG      (CCH / 4)   // 32 k-groups
#define GS      516         // 128*4 + 4 pad (bank spread across k-groups)
#define GSI     68          // 16*4 + 4 pad (padded 16-row weight matrices)

typedef __attribute__((ext_vector_type(2))) float v2f;
typedef __attribute__((ext_vector_type(4))) float v4f;
typedef __attribute__((ext_vector_type(8))) float v8f;

// Native fp32 WMMA: D(16x16,f32) = A(16x4,f32) * B(4x16,f32) + C
__device__ __forceinline__ v8f wmma_k4(v2f a, v2f b, v8f c) {
    return __builtin_amdgcn_wmma_f32_16x16x4_f32(
        /*neg_a=*/false, a, /*neg_b=*/false, b,
        /*c_mod=*/(short)0, c, /*reuse_a=*/false, /*reuse_b=*/false);
}

__device__ __forceinline__ v2f lds_frag(const float* base, int idx) {
    return *(const v2f*)(base + idx);   // idx is even -> 8B aligned -> ds_load_b64
}

// ---------------------------------------------------------------------
// Kernel 1: adapted = ReLU(BN(adapt_w @ f_sem)); f_sins = f_ins + adapted;
//           e_ins = ins_w @ f_sins + ins_b  (-> d_out region 2);
//           sq[n] = sum_c e_ins[c,n]^2     (-> workspace)
// grid = BATCH * (NPTS/TN), block = 256 (8 waves)
// ---------------------------------------------------------------------
__global__ void asis_k1_front(const float* __restrict__ f_sem,
                              const float* __restrict__ f_ins,
                              const float* __restrict__ adapt_w,
                              const float* __restrict__ adapt_b,
                              const float* __restrict__ gamma,
                              const float* __restrict__ beta,
                              const float* __restrict__ mean,
                              const float* __restrict__ var,
                              const float* __restrict__ ins_w,
                              const float* __restrict__ ins_b,
                              float* __restrict__ e_out,
                              float* __restrict__ sq_out) {
    extern __shared__ float smem[];
    float* sWp = smem;                 // adapt_w packed  [KG][GS]
    float* sXp = sWp + KG * GS;        // f_sem / f_sins packed [KG][GS]
    float* sIp = sXp + KG * GS;        // ins_w packed    [KG][GSI]
    float* bnA = sIp + KG * GSI;       // gamma*rsqrt(var+eps)   [128]
    float* bnB = bnA + CCH;            // (b-mean)*bnA + beta    [128]
    float* sib = bnB + CCH;            // ins bias padded        [16]

    const int b  = blockIdx.x / (NPTS / TN);
    const int n0 = (blockIdx.x % (NPTS / TN)) * TN;
    const int t  = threadIdx.x;

    for (int e = t; e < CCH * CCH; e += 256) {          // coalesced read
        int r = e >> 7, c = e & 127;                    // r = out row, c = K
        sWp[(c >> 2) * GS + r * 4 + (c & 3)] = adapt_w[e];
    }
    for (int e = t; e < CCH * TN; e += 256) {
        int c = e >> 7, n = e & 127;                    // c = K, n = point
        sXp[(c >> 2) * GS + n * 4 + (c & 3)] = f_sem[(b * CCH + c) * NPTS + n0 + n];
    }
    for (int e = t; e < 16 * CCH; e += 256) {
        int r = e >> 7, c = e & 127;
        sIp[(c >> 2) * GSI + r * 4 + (c & 3)] = (r < INSO) ? ins_w[r * CCH + c] : 0.0f;
    }
    if (t < CCH) {
        float g = gamma[t] * rsqrtf(var[t] + 1e-5f);
        bnA[t] = g;
        bnB[t] = (adapt_b[t] - mean[t]) * g + beta[t];
    }
    if (t < 16) sib[t] = (t < INSO) ? ins_b[t] : 0.0f;
    __syncthreads();

    const int wave = t >> 5;
    const int lane = t & 31;
    const int lm = lane & 15;   // M (A) / N (B,D) index within tile
    const int lh = lane >> 4;   // K-half select (A,B) / row+8 (D)
    const int r0 = wave * 16;   // output row band of this wave

    v8f zero = {};
    v8f acc[8];
    #pragma unroll
    for (int i = 0; i < 8; ++i) acc[i] = zero;

    const int aoff = (r0 + lm) * 4 + 2 * lh;
    const int boff = lm * 4 + 2 * lh;

    for (int k0 = 0; k0 < KG; ++k0) {
        v2f a = lds_frag(sWp, k0 * GS + aoff);
        #pragma unroll
        for (int nt = 0; nt < 8; ++nt) {
            v2f bb = lds_frag(sXp, k0 * GS + nt * 64 + boff);
            acc[nt] = wmma_k4(a, bb, acc[nt]);
        }
    }
    __syncthreads();   // all reads of sXp (B operand) complete

    // BN + ReLU + add f_ins; write f_sins back (packed layout, row = new K)
    for (int nt = 0; nt < 8; ++nt) {
        int n = nt * 16 + lm;
        #pragma unroll
        for (int m = 0; m < 8; ++m) {
            int row = r0 + m + lh * 8;
            float v = acc[nt][m] * bnA[row] + bnB[row];
            v = fmaxf(v, 0.0f);
            v += f_ins[(b * CCH + row) * NPTS + n0 + n];
            sXp[(row >> 2) * GS + n * 4 + (row & 3)] = v;
        }
    }
    __syncthreads();

    // e_ins head: each wave owns one 16-point column tile
    {
        const int nt = wave;
        v8f a2 = zero;
        for (int k0 = 0; k0 < KG; ++k0) {
            v2f a  = lds_frag(sIp, k0 * GSI + boff);
            v2f bb = lds_frag(sXp, k0 * GS + nt * 64 + boff);
            a2 = wmma_k4(a, bb, a2);
        }
        if (lh == 0) {   // rows 0..7 live here; only 0..4 are real
            int n = n0 + nt * 16 + lm;
            float s = 0.0f;
            #pragma unroll
            for (int m = 0; m < INSO; ++m) {
                float v = a2[m] + sib[m];
                e_out[(b * INSO + m) * NPTS + n] = v;
                s += v * v;
            }
            sq_out[b * NPTS + n] = s;
        }
    }
}

// ---------------------------------------------------------------------
// Kernel 2: per-query top-KNN of neg_dist = 2*<ei,ej> - |ei|^2 - |ej|^2
// grid = BATCH * (NPTS/256), block = 256; one thread per query point.
// e_ins + sq staged interleaved [j][8] so each candidate j costs one
// broadcast ds_load_b128 + ds_load_b64.
// ---------------------------------------------------------------------
__global__ void asis_k2_knn(const float* __restrict__ e_ins,
                            const float* __restrict__ sq,
                            int* __restrict__ nn_idx) {
    extern __shared__ float smem[];
    float* sEJ = smem;   // [NPTS][8] : e0..e4, sq, pad, pad

    const int b  = blockIdx.x / (NPTS / 256);
    const int i0 = (blockIdx.x % (NPTS / 256)) * 256;
    const int t  = threadIdx.x;

    for (int e = t; e < INSO * NPTS; e += 256) {        // coalesced read
        int c = e >> 12, j = e & (NPTS - 1);
        sEJ[j * 8 + c] = e_ins[b * INSO * NPTS + e];
    }
    for (int e = t; e < NPTS; e += 256)
        sEJ[e * 8 + 5] = sq[b * NPTS + e];
    __syncthreads();

    const int i = i0 + t;
    const v4f eA = *(const v4f*)(sEJ + i * 8);        // e0..e3
    const v2f eB = *(const v2f*)(sEJ + i * 8 + 4);    // e4, sqi
    const float sqi = eB.y;

    float best[KNN];
    int   bidx[KNN];
    #pragma unroll
    for (int x = 0; x < KNN; ++x) { best[x] = -3.4e38f; bidx[x] = 0; }
    float kth = -3.4e38f;

    for (int j = 0; j < NPTS; ++j) {
        v4f xA = *(const v4f*)(sEJ + j * 8);          // broadcast b128
        v2f xB = *(const v2f*)(sEJ + j * 8 + 4);      // broadcast b64
        float inner = eA.x * xA.x + eA.y * xA.y + eA.z * xA.z
                    + eA.w * xA.w + eB.x * xB.x;
        float d = 2.0f * inner - sqi - xB.y;
        if (d > kth) {
            int p = KNN - 1;
            while (p > 0 && best[p - 1] < d) {
                best[p] = best[p - 1];
                bidx[p] = bidx[p - 1];
                --p;
            }
            best[p] = d;
            bidx[p] = j;
            kth = best[KNN - 1];
        }
    }
    #pragma unroll
    for (int x = 0; x < KNN; ++x)
        nn_idx[(b * NPTS + i) * KNN + x] = bidx[x];
}

// ---------------------------------------------------------------------
// Kernel 3: f_isem = max_k f_sem[:, nn_idx]; p_sem = sem_w @ f_isem + b
// grid = BATCH * (NPTS/TN), block = 256 (8 waves)
// ---------------------------------------------------------------------
__global__ void asis_k3_sem(const float* __restrict__ f_sem,
                            const int* __restrict__ nn_idx,
                            const float* __restrict__ sem_w,
                            const float* __restrict__ sem_b,
                            float* __restrict__ p_out) {
    extern __shared__ float smem[];
    float* sFp  = smem;                      // f_isem packed [KG][GS]
    float* sWmp = sFp + KG * GS;             // sem_w packed  [KG][GSI]
    float* sB   = sWmp + KG * GSI;           // bias padded [16]
    int*   sIdx = (int*)(sB + 16);           // [TN][KNN]

    const int b  = blockIdx.x / (NPTS / TN);
    const int n0 = (blockIdx.x % (NPTS / TN)) * TN;
    const int t  = threadIdx.x;

    for (int e = t; e < 16 * CCH; e += 256) {
        int r = e >> 7, c = e & 127;
        sWmp[(c >> 2) * GSI + r * 4 + (c & 3)] = (r < SEMO) ? sem_w[r * CCH + c] : 0.0f;
    }
    for (int e = t; e < TN * KNN; e += 256) {
        int n = e / KNN, kk = e % KNN;
        sIdx[n * KNN + kk] = nn_idx[(b * NPTS + n0 + n) * KNN + kk];
    }
    if (t < 16) sB[t] = (t < SEMO) ? sem_b[t] : 0.0f;
    __syncthreads();

    // gather-max over the 30 neighbors (f_sem is L2-resident: 8 MB)
    const float* fb = f_sem + (size_t)b * CCH * NPTS;
    for (int e = t; e < CCH * TN; e += 256) {
        int c = e >> 7, n = e & 127;          // c = channel (K of next GEMM)
        const float* frow = fb + c * NPTS;
        const int* il = sIdx + n * KNN;
        float m = -3.4e38f;
        #pragma unroll 5
        for (int kk = 0; kk < KNN; ++kk) m = fmaxf(m, frow[il[kk]]);
        sFp[(c >> 2) * GS + n * 4 + (c & 3)] = m;
    }
    __syncthreads();

    const int wave = t >> 5;
    const int lane = t & 31;
    const int lm = lane & 15, lh = lane >> 4;
    const int nt = wave;
    const int boff = lm * 4 + 2 * lh;

    v8f acc = {};
    for (int k0 = 0; k0 < KG; ++k0) {
        v2f a  = lds_frag(sWmp, k0 * GSI + boff);
        v2f bb = lds_frag(sFp,  k0 * GS + nt * 64 + boff);
        acc = wmma_k4(a, bb, acc);
    }
    const int n = n0 + nt * 16 + lm;
    #pragma unroll
    for (int m = 0; m < 8; ++m) {
        int row = m + lh * 8;
        if (row < SEMO)
            p_out[(b * SEMO + row) * NPTS + n] = acc[m] + sB[row];
    }
}

// ---------------------------------------------------------------------
extern "C" void kernel_launch(void* const* d_in, const int* in_sizes, int n_in,
                              void* d_out, int out_size, void* d_ws, size_t ws_size,
                              hipStream_t stream) {
    const float* f_sem   = (const float*)d_in[0];
    const float* f_ins   = (const float*)d_in[1];
    const float* adapt_w = (const float*)d_in[2];
    const float* adapt_b = (const float*)d_in[3];
    const float* gamma   = (const float*)d_in[4];
    const float* beta    = (const float*)d_in[5];
    const float* mean    = (const float*)d_in[6];
    const float* var     = (const float*)d_in[7];
    const float* ins_w   = (const float*)d_in[8];
    const float* ins_b   = (const float*)d_in[9];
    const float* sem_w   = (const float*)d_in[10];
    const float* sem_b   = (const float*)d_in[11];
    // d_in[12] is k (==30, baked in as KNN)

    float* p_out = (float*)d_out;                       // [B,13,N]
    float* e_out = p_out + (size_t)BATCH * SEMO * NPTS; // [B,5,N]

    float* sq = (float*)d_ws;                                    // B*N floats
    int*   nn = (int*)((char*)d_ws + (size_t)BATCH * NPTS * 4);  // B*N*KNN ints

    const size_t sm1 = (size_t)(2 * KG * GS + KG * GSI + 2 * CCH + 16) * sizeof(float);
    hipLaunchKernelGGL(asis_k1_front, dim3(BATCH * (NPTS / TN)), dim3(256), sm1, stream,
                       f_sem, f_ins, adapt_w, adapt_b, gamma, beta, mean, var,
                       ins_w, ins_b, e_out, sq);

    const size_t sm2 = (size_t)(NPTS * 8) * sizeof(float);
    hipLaunchKernelGGL(asis_k2_knn, dim3(BATCH * (NPTS / 256)), dim3(256), sm2, stream,
                       e_out, sq, nn);

    const size_t sm3 = (size_t)(KG * GS + KG * GSI + 16) * sizeof(float)
                     + (size_t)TN * KNN * sizeof(int);
    hipLaunchKernelGGL(asis_k3_sem, dim3(BATCH * (NPTS / TN)), dim3(256), sm3, stream,
                       f_sem, nn, sem_w, sem_b, p_out);
}